// OWNNorm_7327214207195
// MI455X (gfx1250) — compile-verified
//
#include <hip/hip_runtime.h>

typedef __attribute__((ext_vector_type(2))) float v2f;
typedef __attribute__((ext_vector_type(8))) float v8f;
typedef unsigned int       u32;
typedef unsigned long long u64;
typedef __attribute__((ext_vector_type(4))) u32 v4u;
typedef __attribute__((ext_vector_type(8))) int v8i;
typedef __attribute__((ext_vector_type(4))) int v4i;

#define FAN   16384
#define NOUT  4096
#define NG    16
#define CROWS 256
#define MSZ   (CROWS * CROWS)
#define NS_ITERS 14

#if defined(__HIP_DEVICE_COMPILE__) && __has_builtin(__builtin_amdgcn_tensor_load_to_lds) && \
    __has_builtin(__builtin_amdgcn_s_wait_tensorcnt)
#define OWN_HAS_TDM 1
#else
#define OWN_HAS_TDM 0
#endif

// ---------------------------------------------------------------------------
// TDM tile copy: 16 rows x 64 fp32 from a row-major tensor (stride in elems)
// into LDS with pitch 68 floats (D# pad: +4 DWORDs after every 64 DWORDs).
// One descriptor per wave; completion tracked with TENSORcnt.
// Fallback: per-lane float4 copy (synchronous, ordered by the block barrier).
// ---------------------------------------------------------------------------
__device__ __forceinline__ void tile_load16x64(const float* gsrc, float* ldst,
                                               int gstrideElems) {
#if OWN_HAS_TDM
    const u64 ga = (u64)(size_t)gsrc;
    const u32 la = (u32)(size_t)ldst;           // LDS byte address (low 32 bits)
    v4u g0;
    g0.x = 1u;                                   // count=1 (valid user descriptor)
    g0.y = la;                                   // lds_addr
    g0.z = (u32)ga;                              // global_addr[31:0]
    g0.w = (u32)((ga >> 32) & 0x01FFFFFFu) | (2u << 30);   // addr[56:32] | type=2
    v8i g1;
    g1[0] = (int)((2u << 16)                     // data_size = 4 bytes
                | (1u << 20)                     // pad_enable
                | (5u << 22)                     // pad_interval: 64 DWORDs
                | (3u << 25));                   // pad_amount:   4 DWORDs
    g1[1] = 0;                                   // tensor_dim0 low16 << 16 (dim0=2^30)
    g1[2] = (int)0x4000u;                        // tensor_dim0 hi | tensor_dim1 low16<<16
    g1[3] = (int)(0x4000u | (64u << 16));        // tensor_dim1 hi | tile_dim0 = 64
    g1[4] = 16;                                  // tile_dim1 = 16 rows, tile_dim2 = 0
    g1[5] = gstrideElems;                        // tensor_dim0_stride[31:0]
    g1[6] = 0;                                   // stride hi | tensor_dim1_stride lo
    g1[7] = 0;
    const v4i z4 = {0, 0, 0, 0};                 // groups 2/3 unused (2D tile)
    const v8i z8 = {0, 0, 0, 0, 0, 0, 0, 0};
    __builtin_amdgcn_tensor_load_to_lds(g0, g1, z4, z4, z8, 0);
#else
    const int lane = (int)(threadIdx.x & 31);
#pragma unroll
    for (int i = 0; i < 8; ++i) {
        const int idx = lane + i * 32;           // 256 float4 slots = 16 rows x 16
        const int r = idx >> 4, c = idx & 15;
        *reinterpret_cast<float4*>(&ldst[r * 68 + c * 4]) =
            *reinterpret_cast<const float4*>(&gsrc[(size_t)r * gstrideElems + c * 4]);
    }
#endif
}

template <int N>
__device__ __forceinline__ void tile_wait() {
#if OWN_HAS_TDM
    __builtin_amdgcn_s_wait_tensorcnt(N);       // immediate required
#endif
}

// ---------------------------------------------------------------------------
// Kernel 1: per-row mean of weight.  grid 4096 x 256
// ---------------------------------------------------------------------------
__global__ void own_row_mean(const float* __restrict__ W, float* __restrict__ mu) {
    __shared__ float red[256];
    const int row = blockIdx.x;
    const int t   = threadIdx.x;
    const float4* Wr = reinterpret_cast<const float4*>(W + (size_t)row * FAN);
    float s = 0.f;
#pragma unroll
    for (int j = 0; j < 16; ++j) {
        float4 v = Wr[j * 256 + t];
        s += (v.x + v.y) + (v.z + v.w);
    }
    red[t] = s;
    __syncthreads();
    for (int off = 128; off > 0; off >>= 1) {
        if (t < off) red[t] += red[t + off];
        __syncthreads();
    }
    if (t == 0) mu[row] = red[0] * (1.0f / FAN);
}

// ---------------------------------------------------------------------------
// Kernel 2: S[g] = Z[g]*Z[g]^T - FAN*mu*mu^T   (centering via rank-1 epilogue)
// grid (4,4,16), block 128.  TDM double-buffered LDS tiles feed f32 WMMA.
// ---------------------------------------------------------------------------
__global__ void own_gram(const float* __restrict__ W, const float* __restrict__ mu,
                         float* __restrict__ S) {
    __shared__ float At[2][64][68];
    __shared__ float Bt[2][64][68];
    const int g    = blockIdx.z;
    const int rb   = blockIdx.y * 64;
    const int cb   = blockIdx.x * 64;
    const int t    = threadIdx.x;
    const int wave = t >> 5;
    const int lane = t & 31;
    const int half = lane >> 4;
    const int lo   = lane & 15;
    const int wrow = wave * 16;

    v8f acc[4] = {v8f{}, v8f{}, v8f{}, v8f{}};

    const float* rowBase = W + (size_t)(g * CROWS + rb + wrow) * FAN;
    const float* colBase = W + (size_t)(g * CROWS + cb + wrow) * FAN;

    // prologue: fill buffer 0 (each wave loads its 16-row quarter)
    tile_load16x64(rowBase, &At[0][wrow][0], FAN);
    tile_load16x64(colBase, &Bt[0][wrow][0], FAN);

    const int NT = FAN / 64;
    for (int kt = 0; kt < NT; ++kt) {
        const int cur = kt & 1;
        if (kt + 1 < NT) {      // uniform branch: prefetch next K-tile via TDM
            tile_load16x64(rowBase + (kt + 1) * 64, &At[cur ^ 1][wrow][0], FAN);
            tile_load16x64(colBase + (kt + 1) * 64, &Bt[cur ^ 1][wrow][0], FAN);
            tile_wait<2>();     // current tile's 2 TDM ops retired (in-order)
        } else {
            tile_wait<0>();
        }
        __syncthreads();        // all waves' quarters visible

        for (int kk = 0; kk < 64; kk += 4) {
            const v2f a = *reinterpret_cast<const v2f*>(&At[cur][wrow + lo][kk + 2 * half]);
#pragma unroll
            for (int j = 0; j < 4; ++j) {
                const v2f b = *reinterpret_cast<const v2f*>(&Bt[cur][j * 16 + lo][kk + 2 * half]);
                acc[j] = __builtin_amdgcn_wmma_f32_16x16x4_f32(
                    false, a, false, b, (short)0, acc[j], false, false);
            }
        }
        __syncthreads();        // done reading cur before it is refilled
    }

    float* Sg = S + (size_t)g * MSZ;
    const float* mug = mu + g * CROWS;
#pragma unroll
    for (int j = 0; j < 4; ++j) {
        const int col = cb + j * 16 + lo;
        const float mc = mug[col] * (float)FAN;
#pragma unroll
        for (int i = 0; i < 8; ++i) {
            const int row = rb + wrow + i + 8 * half;
            Sg[row * CROWS + col] = acc[j][i] - mug[row] * mc;
        }
    }
}

// ---------------------------------------------------------------------------
// Kernel 3: per-group Frobenius norm of S.  grid 16 x 256
// ---------------------------------------------------------------------------
__global__ void own_frob(const float* __restrict__ S, float* __restrict__ f) {
    __shared__ float red[256];
    const int g = blockIdx.x;
    const int t = threadIdx.x;
    const float* Sg = S + (size_t)g * MSZ;
    float s = 0.f;
    for (int j = 0; j < 256; ++j) {
        const float v = Sg[j * 256 + t];
        s += v * v;
    }
    red[t] = s;
    __syncthreads();
    for (int off = 128; off > 0; off >>= 1) {
        if (t < off) red[t] += red[t + off];
        __syncthreads();
    }
    if (t == 0) f[g] = sqrtf(red[0]) + 1e-30f;
}

// ---------------------------------------------------------------------------
// Kernel 4: Y = S / f[g],  Z = I.   grid 4096 x 256
// ---------------------------------------------------------------------------
__global__ void own_ns_init(const float* __restrict__ S, const float* __restrict__ f,
                            float* __restrict__ Y, float* __restrict__ Z) {
    const size_t idx = (size_t)blockIdx.x * blockDim.x + threadIdx.x;
    const int g  = (int)(idx >> 16);
    const int ij = (int)(idx & 65535);
    const int i  = ij >> 8;
    const int j  = ij & 255;
    Y[idx] = S[idx] * (1.0f / f[g]);
    Z[idx] = (i == j) ? 1.0f : 0.0f;
}

// ---------------------------------------------------------------------------
// Kernel 5: batched 256^3 GEMM:  C[g] = alpha*(A[g]@B[g]) + beta*I
// grid (4,4,16), block 128.  (Newton-Schulz workhorse; L2-resident data.)
// ---------------------------------------------------------------------------
__global__ void own_gemm256(const float* __restrict__ A, const float* __restrict__ B,
                            float* __restrict__ C, float alpha, float beta) {
    __shared__ float At[64][68];
    __shared__ float Bt[64][68];
    const int g    = blockIdx.z;
    const int rb   = blockIdx.y * 64;
    const int cb   = blockIdx.x * 64;
    const int t    = threadIdx.x;
    const int wave = t >> 5;
    const int lane = t & 31;
    const int half = lane >> 4;
    const int lo   = lane & 15;
    const int wrow = wave * 16;
    const int tr   = t >> 4;
    const int tc   = t & 15;

    const float* Ag = A + (size_t)g * MSZ;
    const float* Bg = B + (size_t)g * MSZ;

    v8f acc[4] = {v8f{}, v8f{}, v8f{}, v8f{}};

    for (int k0 = 0; k0 < CROWS; k0 += 64) {
#pragma unroll
        for (int p = 0; p < 8; ++p) {
            const int r = p * 8 + tr;
            *reinterpret_cast<float4*>(&At[r][tc * 4]) =
                *reinterpret_cast<const float4*>(&Ag[(rb + r) * CROWS + k0 + tc * 4]);
            *reinterpret_cast<float4*>(&Bt[r][tc * 4]) =
                *reinterpret_cast<const float4*>(&Bg[(k0 + r) * CROWS + cb + tc * 4]);
        }
        __syncthreads();

        for (int kk = 0; kk < 64; kk += 4) {
            const v2f a = *reinterpret_cast<const v2f*>(&At[wrow + lo][kk + 2 * half]);
            const int krow = kk + 2 * half;
#pragma unroll
            for (int j = 0; j < 4; ++j) {
                v2f b;
                b.x = Bt[krow][j * 16 + lo];
                b.y = Bt[krow + 1][j * 16 + lo];
                acc[j] = __builtin_amdgcn_wmma_f32_16x16x4_f32(
                    false, a, false, b, (short)0, acc[j], false, false);
            }
        }
        __syncthreads();
    }

    float* Cg = C + (size_t)g * MSZ;
#pragma unroll
    for (int j = 0; j < 4; ++j) {
        const int col = cb + j * 16 + lo;
#pragma unroll
        for (int i = 0; i < 8; ++i) {
            const int row = rb + wrow + i + 8 * half;
            float v = alpha * acc[j][i];
            if (row == col) v += beta;
            Cg[row * CROWS + col] = v;
        }
    }
}

// ---------------------------------------------------------------------------
// Kernel 6: v[g] = Zinf[g] @ mu[g]  (for the rank-1 centering of the output)
// grid 16 x 256
// ---------------------------------------------------------------------------
__global__ void own_proj_mu(const float* __restrict__ Zf, const float* __restrict__ mu,
                            float* __restrict__ v) {
    const int g = blockIdx.x;
    const int r = threadIdx.x;
    const float* Zr = Zf + (size_t)g * MSZ + (size_t)r * CROWS;
    const float* m  = mu + g * CROWS;
    float s = 0.f;
#pragma unroll 8
    for (int k = 0; k < CROWS; ++k) s += Zr[k] * m[k];
    v[g * CROWS + r] = s;
}

// ---------------------------------------------------------------------------
// Kernel 7: Out[g] = ((Zinf[g] @ Wraw[g]) - v 1^T) * rsqrt(f[g])
// grid (256,4,16), block 128.  TDM double-buffered tiles, K = 256.
// ---------------------------------------------------------------------------
__global__ void own_whiten_apply(const float* __restrict__ Zf, const float* __restrict__ W,
                                 const float* __restrict__ vmu, const float* __restrict__ f,
                                 float* __restrict__ Out) {
    __shared__ float At[2][64][68];   // Zinf tile: rows = output rows, cols = k
    __shared__ float Bt[2][64][68];   // raw W tile: rows = k, cols = output cols
    const int g    = blockIdx.z;
    const int rb   = blockIdx.y * 64;
    const int cb   = blockIdx.x * 64;
    const int t    = threadIdx.x;
    const int wave = t >> 5;
    const int lane = t & 31;
    const int half = lane >> 4;
    const int lo   = lane & 15;
    const int wrow = wave * 16;

    const float* Zg = Zf + (size_t)g * MSZ;

    v8f acc[4] = {v8f{}, v8f{}, v8f{}, v8f{}};

    tile_load16x64(Zg + (size_t)(rb + wrow) * CROWS, &At[0][wrow][0], CROWS);
    tile_load16x64(W + (size_t)(g * CROWS + wrow) * FAN + cb, &Bt[0][wrow][0], FAN);

    const int NT = CROWS / 64;
    for (int kt = 0; kt < NT; ++kt) {
        const int cur = kt & 1;
        const int k0n = (kt + 1) * 64;
        if (kt + 1 < NT) {
            tile_load16x64(Zg + (size_t)(rb + wrow) * CROWS + k0n, &At[cur ^ 1][wrow][0], CROWS);
            tile_load16x64(W + (size_t)(g * CROWS + k0n + wrow) * FAN + cb, &Bt[cur ^ 1][wrow][0], FAN);
            tile_wait<2>();
        } else {
            tile_wait<0>();
        }
        __syncthreads();

        for (int kk = 0; kk < 64; kk += 4) {
            const v2f a = *reinterpret_cast<const v2f*>(&At[cur][wrow + lo][kk + 2 * half]);
            const int krow = kk + 2 * half;
#pragma unroll
            for (int j = 0; j < 4; ++j) {
                v2f b;
                b.x = Bt[cur][krow][j * 16 + lo];
                b.y = Bt[cur][krow + 1][j * 16 + lo];
                acc[j] = __builtin_amdgcn_wmma_f32_16x16x4_f32(
                    false, a, false, b, (short)0, acc[j], false, false);
            }
        }
        __syncthreads();
    }

    const float scale = rsqrtf(f[g]);   // S^{-1/2} = Zinf / sqrt(f)
#pragma unroll
    for (int j = 0; j < 4; ++j) {
        const int col = cb + j * 16 + lo;
#pragma unroll
        for (int i = 0; i < 8; ++i) {
            const int orow = g * CROWS + rb + wrow + i + 8 * half;
            Out[(size_t)orow * FAN + col] = (acc[j][i] - vmu[orow]) * scale;
        }
    }
}

// ---------------------------------------------------------------------------
extern "C" void kernel_launch(void* const* d_in, const int* in_sizes, int n_in,
                              void* d_out, int out_size, void* d_ws, size_t ws_size,
                              hipStream_t stream) {
    (void)in_sizes; (void)n_in; (void)out_size; (void)ws_size;
    const float* W = (const float*)d_in[0];   // [4096,16384] fp32
    float* out = (float*)d_out;               // [4096,16384] fp32
    float* ws  = (float*)d_ws;

    // Workspace (floats): mu[4096], f[16]+pad, v[4096], 5 batched matrices
    float* mu   = ws;                         // 4096
    float* f    = ws + 4096;                  // 16 (+16 pad)
    float* vmu  = ws + 4128;                  // 4096
    float* Sb   = ws + 8224;                  // reused as Y-pong after init
    float* Yb   = Sb + (size_t)NG * MSZ;
    float* Zb   = Yb + (size_t)NG * MSZ;
    float* Tb   = Zb + (size_t)NG * MSZ;
    float* Z2   = Tb + (size_t)NG * MSZ;      // total ~21 MB

    own_row_mean<<<NOUT, 256, 0, stream>>>(W, mu);

    own_gram<<<dim3(4, 4, NG), 128, 0, stream>>>(W, mu, Sb);

    own_frob<<<NG, 256, 0, stream>>>(Sb, f);

    own_ns_init<<<(NG * MSZ) / 256, 256, 0, stream>>>(Sb, f, Yb, Zb);

    // Coupled Newton-Schulz: T = 1.5I - 0.5*Z*Y ; Y <- Y*T ; Z <- T*Z
    float* Yc = Yb; float* Zc = Zb;
    float* Yn = Sb; float* Zn = Z2;
    for (int it = 0; it < NS_ITERS; ++it) {
        own_gemm256<<<dim3(4, 4, NG), 128, 0, stream>>>(Zc, Yc, Tb, -0.5f, 1.5f);
        own_gemm256<<<dim3(4, 4, NG), 128, 0, stream>>>(Yc, Tb, Yn, 1.0f, 0.0f);
        own_gemm256<<<dim3(4, 4, NG), 128, 0, stream>>>(Tb, Zc, Zn, 1.0f, 0.0f);
        float* tmp;
        tmp = Yc; Yc = Yn; Yn = tmp;
        tmp = Zc; Zc = Zn; Zn = tmp;
    }

    own_proj_mu<<<NG, 256, 0, stream>>>(Zc, mu, vmu);

    own_whiten_apply<<<dim3(FAN / 64, 4, NG), 128, 0, stream>>>(Zc, W, vmu, f, out);
}